// IouDistance_24704651886790
// MI455X (gfx1250) — compile-verified
//
#include <hip/hip_runtime.h>
#include <stdint.h>

typedef float v4f __attribute__((ext_vector_type(4)));
typedef float v2f __attribute__((ext_vector_type(2)));

#define EPS_F 1e-16f
#define TILE_M 512          // rhs rows this kernel is specialized for
#define JCHUNK 4            // j-columns per thread
#define JTHREADS (TILE_M / JCHUNK)   // 128
#define ROWT 2              // row-threads per block (256 total threads)
#define ROWS_PER_BLOCK 16
#define KITER (ROWS_PER_BLOCK / ROWT)

// Specialized kernel: M == 512. Stores are the bottleneck (205 MB @ 23.3 TB/s
// ~= 9 us floor); everything else is designed to stay out of the way.
__global__ __launch_bounds__(256) void iou512_kernel(
    const float* __restrict__ lhs,   // [N,2]
    const float* __restrict__ rhs,   // [512,2]
    float* __restrict__ out,         // [N,512]
    int N) {
  // rhs staged in LDS: 512 * (w,h) = 1024 floats = 4 KB.
  __shared__ float s_rhs[2 * TILE_M];

  const int t = threadIdx.x;

  // ---- Async-stage rhs into LDS (gfx1250 ASYNCcnt path) ----
  // Each of the 256 threads copies 16 bytes: global rhs[t*4 .. t*4+3] -> LDS.
#if defined(__gfx1250__)
  {
    // Generic pointer to __shared__ has the LDS byte address in its low 32
    // bits (LDS aperture: addr[63:32] = shared_base, addr[31:0] = LDS offset).
    unsigned ldsAddr = (unsigned)(uint64_t)(void*)&s_rhs[t * 4];
    const float* gsrc = rhs + t * 4;
    asm volatile("global_load_async_to_lds_b128 %0, %1, off"
                 :
                 : "v"(ldsAddr), "v"(gsrc)
                 : "memory");
  }
  asm volatile("s_wait_asynccnt 0" ::: "memory");
#else
  {
    v4f tmp = *(const v4f*)(rhs + t * 4);
    *(v4f*)&s_rhs[t * 4] = tmp;
  }
#endif
  __syncthreads();

  // ---- Pull this thread's 4 rhs boxes into registers (reused for 8 rows) ----
  const int jc = t & (JTHREADS - 1);   // 0..127 : which 4-column chunk
  const int ty = t >> 7;               // 0..1   : row sub-index

  float rw[JCHUNK], rh[JCHUNK], ra[JCHUNK];
  {
    // 32 contiguous bytes of LDS: (w0,h0,w1,h1) (w2,h2,w3,h3)
    v4f p0 = *(const v4f*)&s_rhs[jc * 8 + 0];
    v4f p1 = *(const v4f*)&s_rhs[jc * 8 + 4];
    rw[0] = p0.x; rh[0] = p0.y;
    rw[1] = p0.z; rh[1] = p0.w;
    rw[2] = p1.x; rh[2] = p1.y;
    rw[3] = p1.z; rh[3] = p1.w;
#pragma unroll
    for (int q = 0; q < JCHUNK; ++q) ra[q] = rw[q] * rh[q];
  }

  const int rowBase = blockIdx.x * ROWS_PER_BLOCK;

#pragma unroll
  for (int k = 0; k < KITER; ++k) {
    const int row = rowBase + k * ROWT + ty;
    if (row >= N) break;

    // lhs load is uniform across each wave's 32 lanes -> served by L0 broadcast.
    v2f l = *(const v2f*)(lhs + 2 * (size_t)row);
    const float lw = l.x, lh = l.y;
    const float la = lw * lh;

    v4f o;
#pragma unroll
    for (int q = 0; q < JCHUNK; ++q) {
      float iw = fminf(lw, rw[q]);
      float ih = fminf(lh, rh[q]);
      float inter = iw * ih;
      float uni = la + ra[q] - inter;          // > 0 for uniform(0,1) inputs
      float r = inter * __builtin_amdgcn_rcpf(uni) + EPS_F;
      ((float*)&o)[q] = r;
    }

    // 16B non-temporal store: output (205 MB) is write-once, bigger than L2.
    __builtin_nontemporal_store(
        o, (v4f*)(out + (size_t)row * TILE_M + (size_t)jc * JCHUNK));
  }
}

// Generic fallback for arbitrary M (kept simple; correctness for any shape).
__global__ void iou_generic_kernel(const float* __restrict__ lhs,
                                   const float* __restrict__ rhs,
                                   float* __restrict__ out,
                                   int N, int M) {
  long long idx = (long long)blockIdx.x * blockDim.x + threadIdx.x;
  long long total = (long long)N * M;
  if (idx >= total) return;
  int i = (int)(idx / M);
  int j = (int)(idx % M);
  float lw = lhs[2 * (size_t)i + 0];
  float lh = lhs[2 * (size_t)i + 1];
  float rw = rhs[2 * (size_t)j + 0];
  float rh = rhs[2 * (size_t)j + 1];
  float inter = fminf(lw, rw) * fminf(lh, rh);
  float uni = lw * lh + rw * rh - inter;
  out[idx] = inter * __builtin_amdgcn_rcpf(uni) + EPS_F;
}

extern "C" void kernel_launch(void* const* d_in, const int* in_sizes, int n_in,
                              void* d_out, int out_size, void* d_ws,
                              size_t ws_size, hipStream_t stream) {
  const float* lhs = (const float*)d_in[0];
  const float* rhs = (const float*)d_in[1];
  float* out = (float*)d_out;
  const int N = in_sizes[0] / 2;   // 100000
  const int M = in_sizes[1] / 2;   // 512

  if (M == TILE_M) {
    const int blocks = (N + ROWS_PER_BLOCK - 1) / ROWS_PER_BLOCK;  // 6250
    iou512_kernel<<<blocks, 256, 0, stream>>>(lhs, rhs, out, N);
  } else {
    long long total = (long long)N * M;
    int blocks = (int)((total + 255) / 256);
    iou_generic_kernel<<<blocks, 256, 0, stream>>>(lhs, rhs, out, N, M);
  }
}